// FanBeamReconstruction_80281528696889
// MI455X (gfx1250) — compile-verified
//
#include <hip/hip_runtime.h>

// ---------------- geometry constants (match reference) ----------------
#define NBATCH 16
#define NANG   1024
#define NDET   1024
#define IMG_H  512
#define IMG_W  512
#define VX     1.0f
#define DU     1.25f
#define D_SI   750.0f
#define D_SD   1200.0f
#define PI_F   3.14159265358979323846f

typedef __attribute__((ext_vector_type(2))) float        v2f;
typedef __attribute__((ext_vector_type(8))) float        v8f;
typedef __attribute__((ext_vector_type(4))) unsigned int u32x4;
typedef __attribute__((ext_vector_type(4))) int          i32x4;
typedef __attribute__((ext_vector_type(8))) int          i32x8;

// ---------------------------------------------------------------------
// Tensor Data Mover support (gfx1250). Guarded so compilation never
// breaks: 5-arg builtin on ROCm7.2/clang-22, 6-arg on clang>=23,
// direct-global-load fallback otherwise.
// ---------------------------------------------------------------------
#if defined(__AMDGCN__) && __has_builtin(__builtin_amdgcn_tensor_load_to_lds) && \
    __has_builtin(__builtin_amdgcn_s_wait_tensorcnt)
#define FBP_HAVE_TDM 1
#endif

#ifdef FBP_HAVE_TDM
// Load a tile_rows x tile_cols f32 tile (row stride = row_stride elems)
// from global into LDS (contiguous row-major) via the TDM.
__device__ __forceinline__ void tdm_load_tile_f32(const float* gsrc, unsigned lds_byte,
                                                  unsigned tile_cols, unsigned tile_rows,
                                                  unsigned row_stride) {
    unsigned long long ga = (unsigned long long)(const void*)gsrc;
    // D# group 0: count=1 | lds_addr | global_addr[56:0] | type=2
    u32x4 g0;
    g0[0] = 1u;
    g0[1] = lds_byte;
    g0[2] = (unsigned)ga;
    g0[3] = (unsigned)((ga >> 32) & 0x1FFFFFFu) | (2u << 30);
    // D# group 1: wg_mask=0, data_size=2 (4B), dims/strides/tile dims
    i32x8 g1;
    g1[0] = (int)(2u << 16);                                     // data_size=4B
    g1[1] = (int)((row_stride & 0xFFFFu) << 16);                 // tensor_dim0[15:0]
    g1[2] = (int)((row_stride >> 16) | ((row_stride & 0xFFFFu) << 16)); // dim0 hi | dim1 lo
    g1[3] = (int)((row_stride >> 16) | (tile_cols << 16));       // dim1 hi | tile_dim0
    g1[4] = (int)tile_rows;                                      // tile_dim1 | tile_dim2=0
    g1[5] = (int)row_stride;                                     // dim0_stride[31:0]
    g1[6] = (int)((row_stride & 0xFFFFu) << 16);                 // dim0_stride hi | dim1_stride lo
    g1[7] = (int)(row_stride >> 16);                             // dim1_stride[47:16]
    i32x4 z4 = {0, 0, 0, 0};
#if __clang_major__ >= 23
    i32x8 z8 = {0, 0, 0, 0, 0, 0, 0, 0};
    __builtin_amdgcn_tensor_load_to_lds(g0, g1, z4, z4, z8, 0);
#else
    __builtin_amdgcn_tensor_load_to_lds(g0, g1, z4, z4, 0);
#endif
}
#define TDM_WAIT(n) __builtin_amdgcn_s_wait_tensorcnt(n)
#else
#define TDM_WAIT(n) ((void)0)
#endif

// ---------------------------------------------------------------------
// Kernel 1: h[n] = (1/D) * sum_k filt[k] * cos(2*pi*k*n/D)
// ---------------------------------------------------------------------
__global__ void fbp_build_h(const float* __restrict__ filt, float* __restrict__ h) {
    int n = blockIdx.x * blockDim.x + threadIdx.x;
    if (n >= NDET) return;
    const float w0 = 2.0f * PI_F / (float)NDET;
    float s = 0.0f;
    for (int k = 0; k < NDET; ++k) {
        int ph = (k * n) & (NDET - 1);
        s += filt[k] * cosf(w0 * (float)ph);
    }
    h[n] = s * (1.0f / (float)NDET);
}

// ---------------------------------------------------------------------
// Kernel 2: circulant matrix C[k][n] = h[(n - k) mod D]
// ---------------------------------------------------------------------
__global__ void fbp_build_circ(const float* __restrict__ h, float* __restrict__ C) {
    int idx = blockIdx.x * blockDim.x + threadIdx.x;
    int k = idx >> 10;
    int n = idx & (NDET - 1);
    C[idx] = h[(n - k) & (NDET - 1)];
}

// ---------------------------------------------------------------------
// Kernel 3: per-angle trig tables
// ---------------------------------------------------------------------
__global__ void fbp_trig(const float* __restrict__ angles,
                         float* __restrict__ cosb, float* __restrict__ sinb) {
    int a = blockIdx.x * blockDim.x + threadIdx.x;
    if (a >= NANG) return;
    float b = angles[a];
    cosb[a] = cosf(b);
    sinb[a] = sinf(b);
}

// ---------------------------------------------------------------------
// Kernel 4: F = S @ C via V_WMMA_F32_16X16X4_F32.
// Each wave: 4 M-tiles x 1 N-tile (64x16 output), so one B fragment
// feeds 4 WMMAs. B staged through LDS by the TDM, double buffered on
// TENSORcnt. A streams from global (rows shared by all 8 block waves).
// ---------------------------------------------------------------------
#define MT 4
#define KB 32

__global__ __launch_bounds__(256) void fbp_filter_gemm(
        const float* __restrict__ S,   // [16384 x 1024]
        const float* __restrict__ C,   // [1024 x 1024]
        float* __restrict__ F) {       // [16384 x 1024]
    const int lane = threadIdx.x & 31;
    const int wave = threadIdx.x >> 5;
    const int job  = blockIdx.x * 8 + wave;          // 16384 wave-jobs
    const int tn   = job & 63;                       // N tile
    const int tmg  = job >> 6;                       // 64-row M group
    const int m0   = tmg * (16 * MT);
    const int n0   = tn * 16;
    const int l15  = lane & 15;
    const int khalf = (lane >> 4) << 1;              // 0 or 2

    const float* __restrict__ srow[MT];
#pragma unroll
    for (int t = 0; t < MT; ++t)
        srow[t] = S + (size_t)(m0 + 16 * t + l15) * NDET + khalf;

    v8f acc[MT];
#pragma unroll
    for (int t = 0; t < MT; ++t) acc[t] = (v8f){};

#ifdef FBP_HAVE_TDM
    __shared__ float bstage[8][2][KB * 16];
    const unsigned ldsb[2] = {(unsigned)(size_t)&bstage[wave][0][0],
                              (unsigned)(size_t)&bstage[wave][1][0]};
    tdm_load_tile_f32(C + n0, ldsb[0], 16, KB, NDET);
    for (int kb = 0; kb < NDET / KB; ++kb) {
        if (kb + 1 < NDET / KB) {
            tdm_load_tile_f32(C + (size_t)(kb + 1) * KB * NDET + n0,
                              ldsb[(kb + 1) & 1], 16, KB, NDET);
            TDM_WAIT(1);     // in-order per-wave: block kb now resident
        } else {
            TDM_WAIT(0);
        }
        const float* __restrict__ bb = &bstage[wave][kb & 1][0];
#pragma unroll
        for (int kl = 0; kl < KB; kl += 4) {
            v2f b;
            b.x = bb[(kl + khalf) * 16 + l15];       // ds_load
            b.y = bb[(kl + khalf + 1) * 16 + l15];
            const int k = kb * KB + kl;
#pragma unroll
            for (int t = 0; t < MT; ++t) {
                v2f a = *(const v2f*)(srow[t] + k);  // 8B global load, L2-resident
                acc[t] = __builtin_amdgcn_wmma_f32_16x16x4_f32(
                    false, a, false, b, (short)0, acc[t], false, false);
            }
        }
    }
#else
    const float* __restrict__ ccol = C + (size_t)khalf * NDET + (n0 + l15);
    for (int k = 0; k < NDET; k += 4) {
        v2f b;
        b.x = ccol[(size_t)k * NDET];
        b.y = ccol[(size_t)(k + 1) * NDET];
        __builtin_prefetch(ccol + (size_t)(k + 8) * NDET, 0, 1);
#pragma unroll
        for (int t = 0; t < MT; ++t) {
            v2f a = *(const v2f*)(srow[t] + k);
            acc[t] = __builtin_amdgcn_wmma_f32_16x16x4_f32(
                false, a, false, b, (short)0, acc[t], false, false);
        }
    }
#endif

#pragma unroll
    for (int t = 0; t < MT; ++t) {
        float* __restrict__ out =
            F + (size_t)(m0 + 16 * t + (lane >> 4) * 8) * NDET + n0 + l15;
#pragma unroll
        for (int r = 0; r < 8; ++r)
            out[(size_t)r * NDET] = acc[t][r];
    }
}

// ---------------------------------------------------------------------
// Kernel 5: fan-beam backprojection (VALU-bound; 16 batch accumulators
// amortize geometry; filtered sinogram is L2-resident).
// ---------------------------------------------------------------------
__global__ __launch_bounds__(256) void fbp_backproject(
        const float* __restrict__ F,
        const float* __restrict__ cosb,
        const float* __restrict__ sinb,
        float* __restrict__ out) {
    const int px = blockIdx.x * 16 + (threadIdx.x & 15);
    const int py = blockIdx.y * 16 + (threadIdx.x >> 4);
    const float X = ((float)px - (IMG_W - 1) * 0.5f) * VX;
    const float Y = ((float)py - (IMG_H - 1) * 0.5f) * VX;

    float acc[NBATCH];
#pragma unroll
    for (int b = 0; b < NBATCH; ++b) acc[b] = 0.0f;

    for (int a = 0; a < NANG; ++a) {
        const float c = cosb[a];
        const float s = sinb[a];
        const float t    = D_SI + X * c + Y * s;
        const float invt = 1.0f / t;
        const float u    = D_SD * (Y * c - X * s) * invt;
        const float idx  = u * (1.0f / DU) + (NDET - 1) * 0.5f;
        const bool inside = (idx >= 0.0f) && (idx <= (float)(NDET - 1));
        float fi = floorf(idx);
        fi = fminf(fmaxf(fi, 0.0f), (float)(NDET - 2));
        const int   i0   = (int)fi;
        const float frac = fminf(fmaxf(idx - fi, 0.0f), 1.0f);
        float w = D_SI * invt;
        w = w * w;
        const float wf = inside ? w : 0.0f;
        const float w0 = wf * (1.0f - frac);
        const float w1 = wf * frac;

        const float* __restrict__ p = F + (size_t)a * NDET + i0;
#pragma unroll
        for (int b = 0; b < NBATCH; ++b) {
            acc[b] = fmaf(w0, p[0], fmaf(w1, p[1], acc[b]));
            p += (size_t)NANG * NDET;
        }
    }

    const float scale = PI_F / (float)NANG;
    float* __restrict__ o = out + (size_t)py * IMG_W + px;
#pragma unroll
    for (int b = 0; b < NBATCH; ++b)
        o[(size_t)b * IMG_H * IMG_W] = acc[b] * scale;
}

// ---------------------------------------------------------------------
extern "C" void kernel_launch(void* const* d_in, const int* in_sizes, int n_in,
                              void* d_out, int out_size, void* d_ws, size_t ws_size,
                              hipStream_t stream) {
    const float* sino   = (const float*)d_in[0];  // [16,1024,1024]
    const float* filt   = (const float*)d_in[1];  // [1024]
    const float* angles = (const float*)d_in[2];  // [1024]
    float* out = (float*)d_out;                   // [16,512,512]

    float* wsf  = (float*)d_ws;
    float* h    = wsf;
    float* cosb = wsf + 1024;
    float* sinb = wsf + 2048;
    float* Cm   = wsf + 4096;
    float* F    = Cm + (size_t)NDET * NDET;

    fbp_build_h   <<<NDET / 256, 256, 0, stream>>>(filt, h);
    fbp_build_circ<<<(NDET * NDET) / 256, 256, 0, stream>>>(h, Cm);
    fbp_trig      <<<NANG / 256, 256, 0, stream>>>(angles, cosb, sinb);

    // 16384 wave-jobs (256 M-groups x 64 N-tiles), 8 waves per block
    fbp_filter_gemm<<<16384 / 8, 256, 0, stream>>>(sino, Cm, F);

    fbp_backproject<<<dim3(IMG_W / 16, IMG_H / 16), 256, 0, stream>>>(F, cosb, sinb, out);
}